// GCN_47510928228518
// MI455X (gfx1250) — compile-verified
//
#include <hip/hip_runtime.h>
#include <hip/hip_bf16.h>
#include <math.h>

typedef float v2f __attribute__((ext_vector_type(2)));
typedef float v8f __attribute__((ext_vector_type(8)));

#define TPB 256

// ---------------------------------------------------------------------------
// Utility: fill
// ---------------------------------------------------------------------------
__global__ void fill_k(float* __restrict__ p, float v, int n) {
    int i = blockIdx.x * TPB + threadIdx.x;
    if (i < n) p[i] = v;
}

// ---------------------------------------------------------------------------
// Float atomic max via int/uint ordering trick (init must be very negative).
// Positive floats compare correctly as signed ints (atomicMax on int);
// negative floats compare inversely as unsigned (atomicMin on uint).
// Mixed case resolves correctly because positive-bits < 0x80000000 <= neg-bits.
// ---------------------------------------------------------------------------
__device__ __forceinline__ void atomicMaxFloat(float* addr, float val) {
    if (val >= 0.0f) {
        atomicMax((int*)addr, __float_as_int(val));
    } else {
        atomicMin((unsigned int*)addr, __float_as_uint(val));
    }
}

// ---------------------------------------------------------------------------
// WMMA GEMM: C[nrows x 64] = A[nrows x 64] @ W[64 x 64]  (+= if accumulate)
// f32 via V_WMMA_F32_16X16X4_F32. One wave per 32-row (2x16) tile; W cached
// in LDS with K-pair interleaving so each B fragment is one aligned
// ds_load_b64: wlds[(k>>1)*128 + n*2 + (k&1)] = W[k][n].
// Each B fragment is reused by two A tiles -> 128 WMMA per wave.
// A-frag layout (ISA 7.12.2, 32-bit A 16x4): lanes 0-15 -> M rows, VGPR0/1
//   hold K = 2*half, 2*half+1.  B 4x16 mirrored. C/D: VGPR j = row j + 8*half.
// nrows must be a multiple of 16 (100000 = 6250*16 = 3125*32 here).
// ---------------------------------------------------------------------------
__global__ __launch_bounds__(TPB) void gemm64_wmma(
    const float* __restrict__ A, const float* __restrict__ W,
    float* __restrict__ C, int nrows, int accumulate) {
    __shared__ float wlds[32 * 128];   // K-pair interleaved W
    int tid = threadIdx.x;
    for (int i = tid; i < 64 * 64; i += TPB) {
        int k = i >> 6, n = i & 63;
        wlds[(k >> 1) * 128 + n * 2 + (k & 1)] = W[i];
    }
    __syncthreads();

    int tile = blockIdx.x * (TPB / 32) + (tid >> 5);  // 32-row super-tile
    int r0 = tile * 32;
    if (r0 >= nrows) return;
    bool has2 = (r0 + 16) < nrows;

    int lane = tid & 31;
    int half = lane >> 4;   // selects K pair (A/B) and row+8 (C/D)
    int l16  = lane & 15;

    const float* arow0 = A + (size_t)(r0 + l16) * 64;
    const float* arow1 = has2 ? (arow0 + 16 * 64) : arow0;

    v8f c00 = {}, c01 = {}, c02 = {}, c03 = {};
    v8f c10 = {}, c11 = {}, c12 = {}, c13 = {};

    #pragma unroll
    for (int kk = 0; kk < 16; ++kk) {
        int kb = kk * 4 + half * 2;                    // even K index
        v2f a0 = *(const v2f*)(arow0 + kb);            // global_load_b64
        v2f a1 = *(const v2f*)(arow1 + kb);
        const v2f* wrow = (const v2f*)&wlds[(kb >> 1) * 128];
        v2f b0 = wrow[l16];                            // ds_load_b64, no repack
        v2f b1 = wrow[l16 + 16];
        v2f b2 = wrow[l16 + 32];
        v2f b3 = wrow[l16 + 48];
        c00 = __builtin_amdgcn_wmma_f32_16x16x4_f32(false, a0, false, b0, (short)0, c00, false, false);
        c01 = __builtin_amdgcn_wmma_f32_16x16x4_f32(false, a0, false, b1, (short)0, c01, false, false);
        c02 = __builtin_amdgcn_wmma_f32_16x16x4_f32(false, a0, false, b2, (short)0, c02, false, false);
        c03 = __builtin_amdgcn_wmma_f32_16x16x4_f32(false, a0, false, b3, (short)0, c03, false, false);
        c10 = __builtin_amdgcn_wmma_f32_16x16x4_f32(false, a1, false, b0, (short)0, c10, false, false);
        c11 = __builtin_amdgcn_wmma_f32_16x16x4_f32(false, a1, false, b1, (short)0, c11, false, false);
        c12 = __builtin_amdgcn_wmma_f32_16x16x4_f32(false, a1, false, b2, (short)0, c12, false, false);
        c13 = __builtin_amdgcn_wmma_f32_16x16x4_f32(false, a1, false, b3, (short)0, c13, false, false);
    }

    #pragma unroll
    for (int j = 0; j < 8; ++j) {
        int row = r0 + j + 8 * half;
        float* crow = C + (size_t)row * 64 + l16;
        if (accumulate) {
            crow[0]  += c00[j];
            crow[16] += c01[j];
            crow[32] += c02[j];
            crow[48] += c03[j];
        } else {
            crow[0]  = c00[j];
            crow[16] = c01[j];
            crow[32] = c02[j];
            crow[48] = c03[j];
        }
    }
    if (has2) {
        #pragma unroll
        for (int j = 0; j < 8; ++j) {
            int row = r0 + 16 + j + 8 * half;
            float* crow = C + (size_t)row * 64 + l16;
            if (accumulate) {
                crow[0]  += c10[j];
                crow[16] += c11[j];
                crow[32] += c12[j];
                crow[48] += c13[j];
            } else {
                crow[0]  = c10[j];
                crow[16] = c11[j];
                crow[32] = c12[j];
                crow[48] = c13[j];
            }
        }
    }
}

// ---------------------------------------------------------------------------
// v[d] = sum_h W[d,h] * att[h]   (refactor (x@W)@att == x@(W@att))
// ---------------------------------------------------------------------------
__global__ void attvec_k(const float* __restrict__ w, const float* __restrict__ att,
                         float* __restrict__ v) {
    int d = threadIdx.x;  // 64 threads
    float s = 0.0f;
    for (int h = 0; h < 64; ++h) s += w[d * 64 + h] * att[h];
    v[d] = s;
}

// as_[i] = x[i] . vsrc ; ad_[i] = x[i] . vdst
__global__ void node_logits_k(const float* __restrict__ x,
                              const float* __restrict__ vsrc,
                              const float* __restrict__ vdst,
                              float* __restrict__ as_, float* __restrict__ ad_, int n) {
    int i = blockIdx.x * TPB + threadIdx.x;
    if (i >= n) return;
    const float* xr = x + (size_t)i * 64;
    float s = 0.0f, d = 0.0f;
    for (int h = 0; h < 64; ++h) {
        float xv = xr[h];
        s += xv * vsrc[h];
        d += xv * vdst[h];
    }
    as_[i] = s;
    ad_[i] = d;
}

// ---------------------------------------------------------------------------
// GAT pass 1: per-dst max of leaky_relu(as[src] + ad[dst]) incl. self loops
// ---------------------------------------------------------------------------
__global__ void gat_max_k(const int* __restrict__ ei,
                          const float* __restrict__ as_, const float* __restrict__ ad_,
                          float* __restrict__ mmax, int E, int N) {
    int e = blockIdx.x * TPB + threadIdx.x;
    if (e >= E + N) return;
    int s, d;
    if (e < E) { s = ei[e]; d = ei[E + e]; } else { s = d = e - E; }
    float a = as_[s] + ad_[d];
    a = a > 0.0f ? a : 0.2f * a;
    atomicMaxFloat(&mmax[d], a);
}

// GAT pass 2: z[dst] += exp(a - m[dst])
__global__ void gat_expsum_k(const int* __restrict__ ei,
                             const float* __restrict__ as_, const float* __restrict__ ad_,
                             const float* __restrict__ mmax, float* __restrict__ zsum,
                             int E, int N) {
    int e = blockIdx.x * TPB + threadIdx.x;
    if (e >= E + N) return;
    int s, d;
    if (e < E) { s = ei[e]; d = ei[E + e]; } else { s = d = e - E; }
    float a = as_[s] + ad_[d];
    a = a > 0.0f ? a : 0.2f * a;
    atomicAdd(&zsum[d], expf(a - mmax[d]));
}

// GAT pass 3: out[dst] += alpha * hs[src].  One wave per edge, 2 cols/lane.
__global__ void gat_scatter_k(const int* __restrict__ ei,
                              const float* __restrict__ as_, const float* __restrict__ ad_,
                              const float* __restrict__ mmax, const float* __restrict__ zsum,
                              const float* __restrict__ hs, float* __restrict__ acc,
                              int E, int N) {
    int e = blockIdx.x * (TPB / 32) + (threadIdx.x >> 5);
    if (e >= E + N) return;
    int s, d;
    if (e < E) { s = ei[e]; d = ei[E + e]; } else { s = d = e - E; }
    float a = as_[s] + ad_[d];
    a = a > 0.0f ? a : 0.2f * a;
    float alpha = expf(a - mmax[d]) / zsum[d];
    int lane = threadIdx.x & 31;
    const float* hrow = hs + (size_t)s * 64;
    float* orow = acc + (size_t)d * 64;
    atomicAdd(&orow[lane],      alpha * hrow[lane]);
    atomicAdd(&orow[lane + 32], alpha * hrow[lane + 32]);
}

// ---------------------------------------------------------------------------
// SAGE: agg[dst] += x_src[src]; cnt[dst] += 1.  One wave per edge.
// ---------------------------------------------------------------------------
__global__ void sage_scatter_k(const int* __restrict__ ei, const float* __restrict__ x_src,
                               float* __restrict__ agg, float* __restrict__ cnt, int E) {
    int e = blockIdx.x * (TPB / 32) + (threadIdx.x >> 5);
    if (e >= E) return;
    int s = ei[e], d = ei[E + e];
    int lane = threadIdx.x & 31;
    const float* xr = x_src + (size_t)s * 64;
    float* ar = agg + (size_t)d * 64;
    atomicAdd(&ar[lane],      xr[lane]);
    atomicAdd(&ar[lane + 32], xr[lane + 32]);
    if (lane == 0) atomicAdd(&cnt[d], 1.0f);
}

// agg[i, :] /= max(cnt[i], 1)
__global__ void mean_div_k(float* __restrict__ agg, const float* __restrict__ cnt, int total) {
    int i = blockIdx.x * TPB + threadIdx.x;
    if (i >= total) return;
    float c = cnt[i >> 6];
    agg[i] /= fmaxf(c, 1.0f);
}

// ---------------------------------------------------------------------------
// GCN: deg[dst] += 1 (deg pre-init to 1.0 for self loop)
// ---------------------------------------------------------------------------
__global__ void gcn_deg_k(const int* __restrict__ ei, float* __restrict__ deg, int E) {
    int e = blockIdx.x * TPB + threadIdx.x;
    if (e >= E) return;
    atomicAdd(&deg[ei[E + e]], 1.0f);
}

// GCN scatter: out[dst] += rsqrt(deg[src])*rsqrt(deg[dst]) * xw[src], + self loops
__global__ void gcn_scatter_k(const int* __restrict__ ei, const float* __restrict__ deg,
                              const float* __restrict__ xw, float* __restrict__ acc,
                              int E, int N) {
    int e = blockIdx.x * (TPB / 32) + (threadIdx.x >> 5);
    if (e >= E + N) return;
    int s, d;
    if (e < E) { s = ei[e]; d = ei[E + e]; } else { s = d = e - E; }
    float norm = rsqrtf(deg[s]) * rsqrtf(deg[d]);
    int lane = threadIdx.x & 31;
    const float* xr = xw + (size_t)s * 64;
    float* ar = acc + (size_t)d * 64;
    atomicAdd(&ar[lane],      norm * xr[lane]);
    atomicAdd(&ar[lane + 32], norm * xr[lane + 32]);
}

// ---------------------------------------------------------------------------
// Epilogue: h = relu(acc + bias1 [+ bias2]); out = sigmoid(h @ lin_w + lin_b)
// ---------------------------------------------------------------------------
__global__ void final_out_k(const float* __restrict__ acc,
                            const float* __restrict__ bias1, const float* __restrict__ bias2,
                            const float* __restrict__ lin_w, const float* __restrict__ lin_b,
                            float* __restrict__ out, int n, int out_off) {
    int i = blockIdx.x * TPB + threadIdx.x;
    if (i >= n) return;
    const float* r = acc + (size_t)i * 64;
    float s0 = lin_b[0], s1 = lin_b[1];
    for (int h = 0; h < 64; ++h) {
        float v = r[h] + bias1[h];
        if (bias2) v += bias2[h];
        v = v > 0.0f ? v : 0.0f;
        s0 += v * lin_w[h * 2 + 0];
        s1 += v * lin_w[h * 2 + 1];
    }
    out[(size_t)(out_off + i) * 2 + 0] = 1.0f / (1.0f + expf(-s0));
    out[(size_t)(out_off + i) * 2 + 1] = 1.0f / (1.0f + expf(-s1));
}

// ---------------------------------------------------------------------------
// Host-side orchestration
// ---------------------------------------------------------------------------
extern "C" void kernel_launch(void* const* d_in, const int* in_sizes, int n_in,
                              void* d_out, int out_size, void* d_ws, size_t ws_size,
                              hipStream_t stream) {
    const float* x_m        = (const float*)d_in[0];
    const float* x_j        = (const float*)d_in[1];
    const int*   ei_mm      = (const int*)d_in[2];
    const int*   ei_mj      = (const int*)d_in[3];
    const int*   ei_jj      = (const int*)d_in[4];
    const float* gat_w_src  = (const float*)d_in[5];
    const float* gat_w_dst  = (const float*)d_in[6];
    const float* gat_a_src  = (const float*)d_in[7];
    const float* gat_a_dst  = (const float*)d_in[8];
    const float* gat_bias   = (const float*)d_in[9];
    const float* sage_w_l   = (const float*)d_in[10];
    const float* sage_w_r   = (const float*)d_in[11];
    const float* sage_bias  = (const float*)d_in[12];
    const float* gcn_w      = (const float*)d_in[13];
    const float* gcn_bias   = (const float*)d_in[14];
    const float* lin_w      = (const float*)d_in[15];
    const float* lin_b      = (const float*)d_in[16];
    float*       out        = (float*)d_out;

    const int NM   = in_sizes[0] / 64;
    const int NJ   = in_sizes[1] / 64;
    const int E_mm = in_sizes[2] / 2;
    const int E_mj = in_sizes[3] / 2;
    const int E_jj = in_sizes[4] / 2;

    // --- workspace layout (floats) ---
    float* ws = (float*)d_ws;
    float* hs       = ws;                         // NM*64  GAT source messages
    float* muon_acc = hs + (size_t)NM * 64;       // NM*64  GAT output accum
    float* sage_agg = muon_acc + (size_t)NM * 64; // NJ*64  SAGE sum -> mean
    float* xw       = sage_agg + (size_t)NJ * 64; // NJ*64  x_jets @ gcn_w
    float* jets_acc = xw + (size_t)NJ * 64;       // NJ*64  jets accumulator
    float* as_      = jets_acc + (size_t)NJ * 64; // NM
    float* ad_      = as_ + NM;                   // NM
    float* mmax     = ad_ + NM;                   // NM
    float* zsum     = mmax + NM;                  // NM
    float* cnt      = zsum + NM;                  // NJ
    float* deg      = cnt + NJ;                   // NJ
    float* vsrc     = deg + NJ;                   // 64
    float* vdst     = vsrc + 64;                  // 64

    auto blocks = [](long n) { return (int)((n + TPB - 1) / TPB); };

    // --- init scratch ---
    fill_k<<<blocks((long)NM * 64), TPB, 0, stream>>>(muon_acc, 0.0f, NM * 64);
    fill_k<<<blocks((long)NJ * 64), TPB, 0, stream>>>(sage_agg, 0.0f, NJ * 64);
    fill_k<<<blocks(NJ), TPB, 0, stream>>>(cnt, 0.0f, NJ);
    fill_k<<<blocks(NM), TPB, 0, stream>>>(zsum, 0.0f, NM);
    fill_k<<<blocks(NM), TPB, 0, stream>>>(mmax, -3.0e38f, NM);
    fill_k<<<blocks(NJ), TPB, 0, stream>>>(deg, 1.0f, NJ);  // self-loop degree

    // --- attention vectors: v = W @ att ---
    attvec_k<<<1, 64, 0, stream>>>(gat_w_src, gat_a_src, vsrc);
    attvec_k<<<1, 64, 0, stream>>>(gat_w_dst, gat_a_dst, vdst);
    node_logits_k<<<blocks(NM), TPB, 0, stream>>>(x_m, vsrc, vdst, as_, ad_, NM);

    // --- dense transforms (WMMA GEMMs), 8 waves/block, 32 rows per wave ---
    int gemm_blocks_m = ((NM + 31) / 32 + 7) / 8;
    int gemm_blocks_j = ((NJ + 31) / 32 + 7) / 8;
    gemm64_wmma<<<gemm_blocks_m, TPB, 0, stream>>>(x_m, gat_w_src, hs, NM, 0);
    gemm64_wmma<<<gemm_blocks_j, TPB, 0, stream>>>(x_j, gcn_w, xw, NJ, 0);
    gemm64_wmma<<<gemm_blocks_j, TPB, 0, stream>>>(x_j, sage_w_r, jets_acc, NJ, 0);

    // --- GAT on muons (mm edges + self loops) ---
    int gat_items = E_mm + NM;
    gat_max_k<<<blocks(gat_items), TPB, 0, stream>>>(ei_mm, as_, ad_, mmax, E_mm, NM);
    gat_expsum_k<<<blocks(gat_items), TPB, 0, stream>>>(ei_mm, as_, ad_, mmax, zsum, E_mm, NM);
    gat_scatter_k<<<(gat_items + 7) / 8, TPB, 0, stream>>>(ei_mm, as_, ad_, mmax, zsum,
                                                           hs, muon_acc, E_mm, NM);

    // --- SAGE muons -> jets ---
    sage_scatter_k<<<(E_mj + 7) / 8, TPB, 0, stream>>>(ei_mj, x_m, sage_agg, cnt, E_mj);
    mean_div_k<<<blocks((long)NJ * 64), TPB, 0, stream>>>(sage_agg, cnt, NJ * 64);
    gemm64_wmma<<<gemm_blocks_j, TPB, 0, stream>>>(sage_agg, sage_w_l, jets_acc, NJ, 1);

    // --- GCN on jets (jj edges + self loops) ---
    gcn_deg_k<<<blocks(E_jj), TPB, 0, stream>>>(ei_jj, deg, E_jj);
    int gcn_items = E_jj + NJ;
    gcn_scatter_k<<<(gcn_items + 7) / 8, TPB, 0, stream>>>(ei_jj, deg, xw, jets_acc, E_jj, NJ);

    // --- epilogue: relu + linear + sigmoid ---
    final_out_k<<<blocks(NM), TPB, 0, stream>>>(muon_acc, gat_bias, (const float*)nullptr,
                                                lin_w, lin_b, out, NM, 0);
    final_out_k<<<blocks(NJ), TPB, 0, stream>>>(jets_acc, sage_bias, gcn_bias,
                                                lin_w, lin_b, out, NJ, NM);
}